// GraphSAGE_80066780332145
// MI455X (gfx1250) — compile-verified
//
#include <hip/hip_runtime.h>
#include <hip/hip_bf16.h>

// GraphSAGE (2x SAGEConv, mean aggregation) for MI455X / gfx1250.
// Aggregation = L2-resident f32 atomic scatter; linear layers = fp32 WMMA
// (V_WMMA_F32_16X16X4_F32) with the 1/deg mean folded into the A-fragment load.

typedef __attribute__((ext_vector_type(2))) float v2f;
typedef __attribute__((ext_vector_type(8))) float v8f;

#define D 256

// ---------------- degree ----------------
__global__ __launch_bounds__(256) void sage_deg_kernel(
    const int* __restrict__ dst, float* __restrict__ deg, int E)
{
    int i = blockIdx.x * blockDim.x + threadIdx.x;
    if (i < E) atomicAdd(&deg[dst[i]], 1.0f);
}

__global__ __launch_bounds__(256) void sage_invdeg_kernel(
    float* __restrict__ deg, int Nn)
{
    int i = blockIdx.x * blockDim.x + threadIdx.x;
    if (i < Nn) deg[i] = 1.0f / fmaxf(deg[i], 1.0f);
}

// ---------------- edge scatter (sum of source features onto dst) ----------
// 64 threads per edge, float4 per thread -> coalesced 1KB row moves,
// global_atomic_add_f32 (no return) into the L2-resident accumulator.
__global__ __launch_bounds__(256) void sage_scatter_kernel(
    const float* __restrict__ feat, const int* __restrict__ src,
    const int* __restrict__ dst, float* __restrict__ agg, int E)
{
    long long i = (long long)blockIdx.x * blockDim.x + threadIdx.x;
    int e = (int)(i >> 6);
    if (e >= E) return;
    int q = ((int)i & 63) << 2;                 // float offset within the row
    int s  = src[e];
    int d0 = dst[e];

    // prefetch a future edge's source row segment (global_prefetch_b8)
    int ep = e + 512;
    if (ep < E) __builtin_prefetch(feat + (size_t)src[ep] * D + q, 0, 1);

    const float4 v = *(const float4*)(feat + (size_t)s * D + q);
    float* p = agg + (size_t)d0 * D + q;
    atomicAdd(p + 0, v.x);
    atomicAdd(p + 1, v.y);
    atomicAdd(p + 2, v.z);
    atomicAdd(p + 3, v.w);
}

// ---------------- fused SAGE linear: out = act( (agg*invdeg)@Wl + b + feat@Wr )
// 256 threads = 8 waves; wave tile = 16x16 output; block tile = 32(M) x 64(N).
// fp32 WMMA 16x16x4, K=256 (mean path) + K=256 (self path) into one v8f acc.
__global__ __launch_bounds__(256) void sage_gemm_kernel(
    const float* __restrict__ agg,  const float* __restrict__ invdeg,
    const float* __restrict__ feat, const float* __restrict__ Wl,
    const float* __restrict__ Wr,   const float* __restrict__ bias,
    float* __restrict__ out, int Nn, int do_relu)
{
    const int wave = threadIdx.x >> 5;
    const int lane = threadIdx.x & 31;
    const int half = lane >> 4;      // K-pair selector for A/B fragments
    const int l15  = lane & 15;

    const int m0 = blockIdx.x * 32 + (wave >> 2) * 16;   // output row tile
    const int n0 = blockIdx.y * 64 + (wave & 3) * 16;    // output col tile

    // A-fragment row for this lane (ISA 7.12.2: 32-bit A 16x4 -> M = lane&15)
    int am  = m0 + l15;
    int amc = (am < Nn) ? am : (Nn - 1);                 // clamp for tail tile
    const float sA = invdeg[amc];                        // fold mean division in
    const float* __restrict__ arow = agg  + (size_t)amc * D;
    const float* __restrict__ xrow = feat + (size_t)amc * D;
    const int ncol = n0 + l15;                           // B fragment: N = lane&15

    v8f acc = {};

    // mean-neighbor path: (agg * invdeg) @ Wl
#pragma unroll 4
    for (int k = 0; k < D; k += 4) {
        const int ka = k + 2 * half;                     // lanes 0-15: K0,K1; 16-31: K2,K3
        v2f a, b;
        a.x = arow[ka]     * sA;
        a.y = arow[ka + 1] * sA;
        b.x = Wl[(size_t)(ka)     * D + ncol];
        b.y = Wl[(size_t)(ka + 1) * D + ncol];
        acc = __builtin_amdgcn_wmma_f32_16x16x4_f32(false, a, false, b,
                                                    (short)0, acc, false, false);
    }

    // self path: feat @ Wr
#pragma unroll 4
    for (int k = 0; k < D; k += 4) {
        const int ka = k + 2 * half;
        v2f a, b;
        a.x = xrow[ka];
        a.y = xrow[ka + 1];
        b.x = Wr[(size_t)(ka)     * D + ncol];
        b.y = Wr[(size_t)(ka + 1) * D + ncol];
        acc = __builtin_amdgcn_wmma_f32_16x16x4_f32(false, a, false, b,
                                                    (short)0, acc, false, false);
    }

    const float bn = bias[ncol];
#pragma unroll
    for (int v = 0; v < 8; ++v) {
        // C/D layout: VGPR v -> row v (lanes 0-15) / row v+8 (lanes 16-31)
        const int row = m0 + v + 8 * half;
        if (row < Nn) {
            float val = acc[v] + bn;
            if (do_relu) val = fmaxf(val, 0.0f);
            out[(size_t)row * D + ncol] = val;
        }
    }
}

extern "C" void kernel_launch(void* const* d_in, const int* in_sizes, int n_in,
                              void* d_out, int out_size, void* d_ws, size_t ws_size,
                              hipStream_t stream)
{
    const float* x   = (const float*)d_in[0];
    const int*   ei  = (const int*)d_in[1];
    const float* W1l = (const float*)d_in[2];
    const float* b1  = (const float*)d_in[3];
    const float* W1r = (const float*)d_in[4];
    const float* W2l = (const float*)d_in[5];
    const float* b2  = (const float*)d_in[6];
    const float* W2r = (const float*)d_in[7];
    float* out = (float*)d_out;

    const int Nn = in_sizes[0] / D;       // 50000
    const int E  = in_sizes[1] / 2;       // 800000
    const int* src = ei;
    const int* dst = ei + E;

    // workspace layout: agg[Nn*D] | deg[Nn] | h[Nn*D]
    float* agg = (float*)d_ws;
    float* deg = agg + (size_t)Nn * D;
    float* h   = deg + (size_t)Nn;

    const size_t aggBytes = (size_t)Nn * D * sizeof(float);
    const size_t degBytes = (size_t)Nn * sizeof(float);

    const int blk = 256;
    const long long scatterThreads = (long long)E * 64;
    const int scatterBlocks = (int)((scatterThreads + blk - 1) / blk);
    dim3 gemmGrid((Nn + 31) / 32, D / 64);

    // degrees (shared by both layers)
    hipMemsetAsync(agg, 0, aggBytes + degBytes, stream);
    sage_deg_kernel<<<(E + blk - 1) / blk, blk, 0, stream>>>(dst, deg, E);
    sage_invdeg_kernel<<<(Nn + blk - 1) / blk, blk, 0, stream>>>(deg, Nn);

    // layer 1: h = relu(mean(x) @ W1l + b1 + x @ W1r)
    sage_scatter_kernel<<<scatterBlocks, blk, 0, stream>>>(x, src, dst, agg, E);
    sage_gemm_kernel<<<gemmGrid, blk, 0, stream>>>(agg, deg, x, W1l, W1r, b1,
                                                   h, Nn, /*relu=*/1);

    // layer 2: out = mean(h) @ W2l + b2 + h @ W2r
    hipMemsetAsync(agg, 0, aggBytes, stream);
    sage_scatter_kernel<<<scatterBlocks, blk, 0, stream>>>(h, src, dst, agg, E);
    sage_gemm_kernel<<<gemmGrid, blk, 0, stream>>>(agg, deg, h, W2l, W2r, b2,
                                                   out, Nn, /*relu=*/0);
}